// GNNEncoder_53558242181513
// MI455X (gfx1250) — compile-verified
//
#include <hip/hip_runtime.h>

// ---------------------------------------------------------------------------
// Hetero-GraphSAGE encoder for MI455X (gfx1250, wave32, WMMA).
// fp32 end-to-end using V_WMMA_F32_16X16X4_F32 (chip's fp32 matrix path).
// A-tiles staged with GLOBAL_LOAD_ASYNC_TO_LDS_B128 (double-buffered) when the
// toolchain declares the builtin; B-tiles k-pair-packed in LDS so every B
// fragment is one aligned b64 load (no v_mov shuffles in the inner loop).
// ---------------------------------------------------------------------------

typedef __attribute__((ext_vector_type(2))) float v2f;
typedef __attribute__((ext_vector_type(8))) float v8f;

#if defined(__HIP_DEVICE_COMPILE__) && \
    __has_builtin(__builtin_amdgcn_global_load_async_to_lds_b128) && \
    __has_builtin(__builtin_amdgcn_s_wait_asynccnt)
#define HAVE_ASYNC 1
// builtin signature (from hipcc diagnostic): param0 = v4i __device__* (AS1),
// param1 = LDS pointer, params 2/3 = imm offset, imm cpol.
typedef int v4i_ld __attribute__((vector_size(4 * sizeof(int))));
typedef __attribute__((address_space(1))) v4i_ld as1_v4i;
typedef __attribute__((address_space(3))) v4i_ld as3_v4i;
#else
#define HAVE_ASYNC 0
#endif

#define N_NEWS 100000
#define N_KW   20000
#define N_ST   3000
#define E_LK   320000
#define E_MS   160000
#define FEAT   768
#define HID    256

// ---------------------------------------------------------------- utilities

__global__ void k_zero(float* __restrict__ p, long long n) {
    long long i = (long long)blockIdx.x * blockDim.x + threadIdx.x;
    long long stride = (long long)gridDim.x * blockDim.x;
    for (; i < n; i += stride) p[i] = 0.0f;
}

// out[row, :] = emb[ids[row], :]   (blockDim.x == HID)
__global__ void k_gather(const float* __restrict__ emb, const int* __restrict__ ids,
                         float* __restrict__ out, int n) {
    int row = blockIdx.x;
    if (row >= n) return;
    int s = ids[row];
    out[(long long)row * HID + threadIdx.x] = emb[(long long)s * HID + threadIdx.x];
}

// cnt[dst[e]] += 1
__global__ void k_count(const int* __restrict__ dst, int E, float* __restrict__ cnt) {
    int e = blockIdx.x * blockDim.x + threadIdx.x;
    if (e < E) atomicAdd(&cnt[dst[e]], 1.0f);
}

// agg[dst[e], :] += x[src[e], :]   (one block per edge, blockDim.x == HID)
__global__ void k_scatter(const float* __restrict__ x, const int* __restrict__ src,
                          const int* __restrict__ dst, int E, float* __restrict__ agg) {
    int e = blockIdx.x;
    if (e >= E) return;
    int s = src[e], d = dst[e];
    atomicAdd(&agg[(long long)d * HID + threadIdx.x],
              x[(long long)s * HID + threadIdx.x]);
}

// ----------------------------------------------------------- WMMA fp32 GEMM
// D[M x 256] (+)= act( (A[M x K] @ W[K x 256]) * rowscale (+ bias) (+ D) )
//   rscale : optional; output row m scaled by 1/max(rscale[m],1).  By linearity
//            (sum @ Wl)/cnt == (sum/cnt) @ Wl, so the mean division is fused
//            into the epilogue and the A copy stays a straight memcpy (async).
//   accum  : D_old added in the epilogue (after scaling, before relu)
//   relu   : fmaxf(.,0) on store
// Block: 256 thr = 8 wave32.  Tile: BM=128 x BN=64, BK=32, double-buffered.
// Wave w owns rows [w*16, w*16+16), 4 accumulators cover the 64 N columns.
// WMMA fragment layouts per CDNA5 ISA 7.12.2 (32-bit A 16x4, 32-bit C/D 16x16).

#define BM 128
#define BN 64
#define BK 32
#define LDA 36    // 144B row stride: 16B aligned, conflict-free strided b64 reads
#define LDB2 160  // packed-B row stride (floats): 640B, 8B aligned, +32-bank skew

__global__ __launch_bounds__(256)
void k_gemm_wmma_f32(const float* __restrict__ A, const float* __restrict__ W,
                     const float* __restrict__ bias, const float* __restrict__ rscale,
                     float* __restrict__ D, int M, int K,
                     int addBias, int accum, int doRelu) {
    // double-buffered tiles
    __shared__ float As[2][BM][LDA];        // A: row-major, padded
    __shared__ float Bs[2][BK / 2][LDB2];   // B: k-pair packed: (k,n) -> [k/2][2n+(k&1)]

    const int tid  = threadIdx.x;
    const int wave = tid >> 5;
    const int lane = tid & 31;
    const int hi   = lane >> 4;   // half-wave select
    const int lo   = lane & 15;
    const int m_block = blockIdx.x * BM;
    const int n_block = blockIdx.y * BN;

    v8f acc[4];
    #pragma unroll
    for (int ns = 0; ns < 4; ++ns)
        #pragma unroll
        for (int j = 0; j < 8; ++j) acc[ns][j] = 0.0f;

    // ---- tile staging: A is a straight 16B/lane copy (async when available);
    //      B is loaded through registers and k-pair packed into LDS.
    auto stage = [&](int k0, int buf) {
        #pragma unroll
        for (int i = 0; i < 4; ++i) {           // A: 128x32 = 1024 float4
            int lin = tid + i * 256;
            int row = lin >> 3;
            int c4  = (lin & 7) << 2;
            int gr  = m_block + row;
            if (gr >= M) gr = M - 1;            // clamp: OOB rows never stored
            const float* gp = A + (long long)gr * K + k0 + c4;
            float*       lp = &As[buf][row][c4];
#if HAVE_ASYNC
            __builtin_amdgcn_global_load_async_to_lds_b128(
                (as1_v4i*)gp, (as3_v4i*)lp, 0, 0);
#else
            if (k0 + BK < K)                    // gfx1250 global_prefetch_b8
                __builtin_prefetch(gp + BK, 0, 1);
            *reinterpret_cast<float4*>(lp) = *reinterpret_cast<const float4*>(gp);
#endif
        }
        // B: 32x64 tile as 16 k-pairs x 64 columns; one (kpair, 4-col) unit/thread
        int kp = tid >> 4;
        int n4 = (tid & 15) << 2;
        float4 r0 = *reinterpret_cast<const float4*>(
            W + (long long)(k0 + 2 * kp    ) * HID + n_block + n4);
        float4 r1 = *reinterpret_cast<const float4*>(
            W + (long long)(k0 + 2 * kp + 1) * HID + n_block + n4);
        float2* bp = reinterpret_cast<float2*>(&Bs[buf][kp][0]);
        bp[n4 + 0] = make_float2(r0.x, r1.x);
        bp[n4 + 1] = make_float2(r0.y, r1.y);
        bp[n4 + 2] = make_float2(r0.z, r1.z);
        bp[n4 + 3] = make_float2(r0.w, r1.w);
    };

    const int ntile = K / BK;
    stage(0, 0);
#if HAVE_ASYNC
    __builtin_amdgcn_s_wait_asynccnt(0);
#endif
    __syncthreads();

    for (int t = 0; t < ntile; ++t) {
        const int buf = t & 1;
        if (t + 1 < ntile) stage((t + 1) * BK, buf ^ 1);

        // --- 8 k-steps of V_WMMA_F32_16X16X4_F32, 4 N-subtiles each
        #pragma unroll
        for (int kk = 0; kk < BK; kk += 4) {
            // A 16x4 f32: lanes 0-15 -> K=kk,kk+1 ; lanes 16-31 -> K=kk+2,kk+3
            v2f a = *reinterpret_cast<const v2f*>(&As[buf][wave * 16 + lo][kk + hi * 2]);
            #pragma unroll
            for (int ns = 0; ns < 4; ++ns) {
                // packed pair (kk+2*hi, kk+2*hi+1) at [kk/2+hi][2*bn] -> one b64
                v2f b = *reinterpret_cast<const v2f*>(
                    &Bs[buf][(kk >> 1) + hi][(ns * 16 + lo) << 1]);
                acc[ns] = __builtin_amdgcn_wmma_f32_16x16x4_f32(
                    /*neg_a=*/false, a, /*neg_b=*/false, b,
                    /*c_mod=*/(short)0, acc[ns],
                    /*reuse_a=*/false, /*reuse_b=*/false);
            }
        }
#if HAVE_ASYNC
        __builtin_amdgcn_s_wait_asynccnt(0);    // next A tile landed in LDS
#endif
        __syncthreads();                        // publish next tile / retire this one
    }

    // --- epilogue: C/D layout -> lane lo = column, VGPR j = row j (+8 for hi half)
    float inv[8];
    #pragma unroll
    for (int j = 0; j < 8; ++j) {
        int gr = m_block + wave * 16 + j + hi * 8;
        inv[j] = 1.0f;
        if (rscale && gr < M) inv[j] = 1.0f / fmaxf(rscale[gr], 1.0f);
    }
    #pragma unroll
    for (int ns = 0; ns < 4; ++ns)
        #pragma unroll
        for (int j = 0; j < 8; ++j) {
            int gr = m_block + wave * 16 + j + hi * 8;
            if (gr >= M) continue;
            int gc = n_block + ns * 16 + lo;
            float v = acc[ns][j] * inv[j];
            if (addBias) v += bias[gc];
            if (accum)   v += D[(long long)gr * HID + gc];
            if (doRelu)  v = fmaxf(v, 0.0f);
            D[(long long)gr * HID + gc] = v;
        }
}

// ------------------------------------------------------------- host helpers

static inline void launch_zero(float* p, long long n, hipStream_t st) {
    hipLaunchKernelGGL(k_zero, dim3(2048), dim3(256), 0, st, p, n);
}

static inline void launch_gemm(const float* A, const float* W, const float* bias,
                               const float* rscale, float* D, int M, int K,
                               int addBias, int accum, int relu, hipStream_t st) {
    dim3 grid((M + BM - 1) / BM, HID / BN);
    hipLaunchKernelGGL(k_gemm_wmma_f32, grid, dim3(256), 0, st,
                       A, W, bias, rscale, D, M, K, addBias, accum, relu);
}

// One hetero layer: relations 0=news->kw, 1=kw->news, 2=news->st, 3=st->news
static void run_layer(const float* hn, const float* hk, const float* hs,
                      const float* Wl, const float* bl, const float* Wr,
                      const int* elks, const int* elkd,
                      const int* emss, const int* emsd,
                      const float* cnt_kw, const float* cnt_n_lk,
                      const float* cnt_n_ms, const float* cnt_st,
                      float* agg, float* on, float* ok, float* os,
                      int relu, hipStream_t st) {
    const long long WW = (long long)HID * HID;
    // --- out_kw = mean_{news->kw}(h_news) @ Wl0 + bl0 + h_kw @ Wr0
    launch_zero(agg, (long long)N_KW * HID, st);
    hipLaunchKernelGGL(k_scatter, dim3(E_LK), dim3(HID), 0, st, hn, elks, elkd, E_LK, agg);
    launch_gemm(agg, Wl + 0 * WW, bl + 0 * HID, cnt_kw, ok, N_KW, HID, 1, 0, 0, st);
    launch_gemm(hk,  Wr + 0 * WW, nullptr,      nullptr, ok, N_KW, HID, 0, 1, relu, st);
    // --- out_news = mean_{kw->news}(h_kw)@Wl1 + bl1 + h_news@Wr1
    //              + mean_{st->news}(h_st)@Wl3 + bl3 + h_news@Wr3
    launch_zero(agg, (long long)N_NEWS * HID, st);
    hipLaunchKernelGGL(k_scatter, dim3(E_LK), dim3(HID), 0, st, hk, elkd, elks, E_LK, agg);
    launch_gemm(agg, Wl + 1 * WW, bl + 1 * HID, cnt_n_lk, on, N_NEWS, HID, 1, 0, 0, st);
    launch_gemm(hn,  Wr + 1 * WW, nullptr,      nullptr,  on, N_NEWS, HID, 0, 1, 0, st);
    launch_zero(agg, (long long)N_NEWS * HID, st);
    hipLaunchKernelGGL(k_scatter, dim3(E_MS), dim3(HID), 0, st, hs, emsd, emss, E_MS, agg);
    launch_gemm(agg, Wl + 3 * WW, bl + 3 * HID, cnt_n_ms, on, N_NEWS, HID, 1, 1, 0, st);
    launch_gemm(hn,  Wr + 3 * WW, nullptr,      nullptr,  on, N_NEWS, HID, 0, 1, relu, st);
    // --- out_st = mean_{news->st}(h_news) @ Wl2 + bl2 + h_st @ Wr2
    launch_zero(agg, (long long)N_ST * HID, st);
    hipLaunchKernelGGL(k_scatter, dim3(E_MS), dim3(HID), 0, st, hn, emss, emsd, E_MS, agg);
    launch_gemm(agg, Wl + 2 * WW, bl + 2 * HID, cnt_st, os, N_ST, HID, 1, 0, 0, st);
    launch_gemm(hs,  Wr + 2 * WW, nullptr,      nullptr, os, N_ST, HID, 0, 1, relu, st);
}

// ---------------------------------------------------------------- entry

extern "C" void kernel_launch(void* const* d_in, const int* in_sizes, int n_in,
                              void* d_out, int out_size, void* d_ws, size_t ws_size,
                              hipStream_t stream) {
    (void)in_sizes; (void)n_in; (void)out_size; (void)ws_size;

    const float* x_news   = (const float*)d_in[0];
    const int*   kw_ids   = (const int*)d_in[1];
    const int*   st_ids   = (const int*)d_in[2];
    const int*   e_lk_src = (const int*)d_in[3];
    const int*   e_lk_dst = (const int*)d_in[4];
    const int*   e_ms_src = (const int*)d_in[5];
    const int*   e_ms_dst = (const int*)d_in[6];
    const float* news_W   = (const float*)d_in[7];
    const float* news_b   = (const float*)d_in[8];
    const float* kw_emb   = (const float*)d_in[9];
    const float* st_emb   = (const float*)d_in[10];
    const float* c1_Wl    = (const float*)d_in[11];
    const float* c1_bl    = (const float*)d_in[12];
    const float* c1_Wr    = (const float*)d_in[13];
    const float* c2_Wl    = (const float*)d_in[14];
    const float* c2_bl    = (const float*)d_in[15];
    const float* c2_Wr    = (const float*)d_in[16];

    // ---- workspace carve-up (floats)
    float* ws = (float*)d_ws;
    float* h_news = ws;  ws += (long long)N_NEWS * HID;
    float* t_news = ws;  ws += (long long)N_NEWS * HID;
    float* h_kw   = ws;  ws += (long long)N_KW * HID;
    float* t_kw   = ws;  ws += (long long)N_KW * HID;
    float* h_st   = ws;  ws += (long long)N_ST * HID;
    float* t_st   = ws;  ws += (long long)N_ST * HID;
    float* agg    = ws;  ws += (long long)N_NEWS * HID;
    float* cnt_kw   = ws; ws += N_KW;
    float* cnt_n_lk = ws; ws += N_NEWS;
    float* cnt_n_ms = ws; ws += N_NEWS;
    float* cnt_st   = ws; ws += N_ST;

    float* o_news = (float*)d_out;
    float* o_kw   = o_news + (long long)N_NEWS * HID;
    float* o_st   = o_kw   + (long long)N_KW * HID;

    // ---- input encoders
    // h_news = relu(x_news @ news_W + news_b)   [100000x768]x[768x256]
    launch_gemm(x_news, news_W, news_b, nullptr, h_news, N_NEWS, FEAT, 1, 0, 1, stream);
    hipLaunchKernelGGL(k_gather, dim3(N_KW), dim3(HID), 0, stream, kw_emb, kw_ids, h_kw, N_KW);
    hipLaunchKernelGGL(k_gather, dim3(N_ST), dim3(HID), 0, stream, st_emb, st_ids, h_st, N_ST);

    // ---- degree counts (shared by both layers)
    launch_zero(cnt_kw,   N_KW,   stream);
    launch_zero(cnt_n_lk, N_NEWS, stream);
    launch_zero(cnt_n_ms, N_NEWS, stream);
    launch_zero(cnt_st,   N_ST,   stream);
    hipLaunchKernelGGL(k_count, dim3((E_LK + 255) / 256), dim3(256), 0, stream, e_lk_dst, E_LK, cnt_kw);
    hipLaunchKernelGGL(k_count, dim3((E_LK + 255) / 256), dim3(256), 0, stream, e_lk_src, E_LK, cnt_n_lk);
    hipLaunchKernelGGL(k_count, dim3((E_MS + 255) / 256), dim3(256), 0, stream, e_ms_src, E_MS, cnt_n_ms);
    hipLaunchKernelGGL(k_count, dim3((E_MS + 255) / 256), dim3(256), 0, stream, e_ms_dst, E_MS, cnt_st);

    // ---- layer 1 (ReLU fused into final GEMM of each output)
    run_layer(h_news, h_kw, h_st, c1_Wl, c1_bl, c1_Wr,
              e_lk_src, e_lk_dst, e_ms_src, e_ms_dst,
              cnt_kw, cnt_n_lk, cnt_n_ms, cnt_st,
              agg, t_news, t_kw, t_st, /*relu=*/1, stream);

    // ---- layer 2 (no activation), straight into d_out
    run_layer(t_news, t_kw, t_st, c2_Wl, c2_bl, c2_Wr,
              e_lk_src, e_lk_dst, e_ms_src, e_ms_dst,
              cnt_kw, cnt_n_lk, cnt_n_ms, cnt_st,
              agg, o_news, o_kw, o_st, /*relu=*/0, stream);
}